// Describe_6047313952809
// MI455X (gfx1250) — compile-verified
//
#include <hip/hip_runtime.h>
#include <hip/hip_bf16.h>
#include <stdint.h>

typedef __attribute__((ext_vector_type(2))) float v2f;
typedef __attribute__((ext_vector_type(8))) float v8f;

#define NB 256     // batch
#define NC 2048    // channels
#define NHW 196    // spatial (49 float4)
#define NE 16      // experts
#define NA 3000    // answers
#define NROWS 512  // padded gathered rows (<= 256 + 15*16)
#define NTILES 32  // NROWS/16
#define LPITCH 36  // LDS pitch in words: 144B = 9*16B (b128-aligned, conflict-free)

// ---------------- Kernel 0: deterministic expert bucketing (1 thread) ----------
__global__ void bucket_kernel(const int* __restrict__ inst,
                              int* __restrict__ rowToSample,
                              int* __restrict__ tileExpert) {
    if (blockIdx.x != 0 || threadIdx.x != 0) return;
    for (int t = 0; t < NTILES; ++t) tileExpert[t] = -1;
    int row = 0;
    for (int e = 0; e < NE; ++e) {
        int start = row;
        for (int b = 0; b < NB; ++b)
            if ((inst[b] & 15) == e) rowToSample[row++] = b;
        if (row > start) {
            while (row & 15) rowToSample[row++] = -1;   // pad tile
            for (int t = start >> 4; t < (row >> 4); ++t) tileExpert[t] = e;
        }
    }
    while (row < NROWS) rowToSample[row++] = -1;
}

// ---------------- Kernel 1: masked-average pool, gathered into Xg --------------
__global__ void pool_kernel(const float* __restrict__ mask,
                            const float* __restrict__ feat,
                            const int* __restrict__ rowToSample,
                            float* __restrict__ Xg) {
    __shared__ alignas(16) float wm[224];   // 56 float4, zero-padded past 196
    __shared__ float red[256];
    const int row = blockIdx.x;
    const int tid = threadIdx.x;
    const int s = rowToSample[row];
    if (s < 0) {   // padded row -> zeros so GEMM is safe
        for (int c = tid; c < NC; c += 256) Xg[row * NC + c] = 0.0f;
        return;
    }
    float mv = 0.0f;
    if (tid < NHW) mv = mask[s * NHW + tid] + 1e-10f;
    red[tid] = mv;
    __syncthreads();
    for (int off = 128; off > 0; off >>= 1) {
        if (tid < off) red[tid] += red[tid + off];
        __syncthreads();
    }
    const float inv = 1.0f / red[0];
    if (tid < 224) wm[tid] = (tid < NHW) ? mv * inv : 0.0f;
    __syncthreads();

    const int lane = tid & 31, wv = tid >> 5;
    const float4* wm4 = (const float4*)wm;
    const float4* fbase4 = (const float4*)(feat + (size_t)s * NC * NHW);
    for (int c = wv; c < NC; c += 8) {
        const float4* fc4 = fbase4 + c * (NHW / 4);
        float sum = 0.0f;
        for (int i = lane; i < NHW / 4; i += 32) {   // 49 float4 per channel
            float4 f = fc4[i];
            float4 w = wm4[i];
            sum += f.x * w.x + f.y * w.y + f.z * w.z + f.w * w.w;
        }
        #pragma unroll
        for (int off = 16; off > 0; off >>= 1) sum += __shfl_down(sum, off, 32);
        if (lane == 0) Xg[row * NC + c] = sum;
    }
}

// ---------------- Kernel 2: expert-grouped GEMM via V_WMMA_F32_16X16X4_F32 -----
// Block: 128 threads = 4 waves. Block tile: 16(M) x 128(N); each wave owns 16x32
// (two v8f accumulators -> A fragment reused by 2 WMMAs). K blocked by 32.
// Staging: double-buffered LDS filled with GLOBAL_LOAD_ASYNC_TO_LDS_B128
// (ASYNCcnt-tracked, no VGPR round-trip), overlapped with WMMA compute.
__global__ void gemm_kernel(const float* __restrict__ Xg,
                            const float* __restrict__ W,
                            const float* __restrict__ bias,
                            const int* __restrict__ rowToSample,
                            const int* __restrict__ tileExpert,
                            float* __restrict__ out) {
    const int mt = blockIdx.y;
    const int e = tileExpert[mt];
    if (e < 0) return;                       // block-uniform early out

    __shared__ alignas(16) float Xs[2][16 * LPITCH];    // 2 x  2304 B
    __shared__ alignas(16) float Ws[2][128 * LPITCH];   // 2 x 18432 B

    const int tid = threadIdx.x;
    const int lane = tid & 31;
    const int wv = tid >> 5;
    const int n0 = blockIdx.x * 128;
    const int m0 = mt * 16;
    const int nIdx0 = n0 + wv * 32 + (lane & 15);
    const int nIdx1 = nIdx0 + 16;
    const float* Wbase = W + (size_t)e * NA * NC;

    // clamp instead of branch: values for n >= NA are computed but never stored
    const int nb0 = nIdx0 < NA ? nIdx0 : NA - 1;
    const int nb1 = nIdx1 < NA ? nIdx1 : NA - 1;
    const float bval0 = bias[e * NA + nb0];
    const float bval1 = bias[e * NA + nb1];
    v8f acc0, acc1;
    #pragma unroll
    for (int i = 0; i < 8; ++i) { acc0[i] = bval0; acc1[i] = bval1; }

    const int khalf = (lane >> 4) * 2;                        // 0 or 2
    const int arow  = (lane & 15) * LPITCH;                   // A fragment row (M)
    const int brow0 = (wv * 32 + (lane & 15)) * LPITCH;       // B fragment rows (N)
    const int brow1 = brow0 + 16 * LPITCH;

    // staging coordinates (float4 granularity; 8 float4 per 32-wide row)
    const int sr = tid >> 3;          // row 0..15
    const int sc = (tid & 7) * 4;     // col 0,4,..,28

    // async fill of buffer `buf` with K-block starting at kb (9 b128 per thread)
    auto issue = [&](int kb, int buf) {
        {
            uint32_t xlds = (uint32_t)(uintptr_t)&Xs[buf][sr * LPITCH + sc];
            const float* xg = &Xg[(m0 + sr) * NC + kb + sc];
            asm volatile("global_load_async_to_lds_b128 %0, %1, off"
                         :: "v"(xlds), "v"(xg) : "memory");
        }
        #pragma unroll
        for (int j = 0; j < 8; ++j) {
            int r = sr + j * 16;
            int a = n0 + r;
            a = a < NA ? a : NA - 1;               // clamp: no divergence
            uint32_t wlds = (uint32_t)(uintptr_t)&Ws[buf][r * LPITCH + sc];
            const float* wp = &Wbase[(size_t)a * NC + kb + sc];
            asm volatile("global_load_async_to_lds_b128 %0, %1, off"
                         :: "v"(wlds), "v"(wp) : "memory");
        }
    };

    issue(0, 0);
    int cur = 0;
    for (int kb = 0; kb < NC; kb += 32) {
        if (kb + 32 < NC) {
            issue(kb + 32, cur ^ 1);              // prefetch next K-block
            // async loads complete in order: <=9 outstanding means buffer
            // `cur`'s 9 loads from the previous issue are done
            asm volatile("s_wait_asynccnt 0x9" ::: "memory");
        } else {
            asm volatile("s_wait_asynccnt 0x0" ::: "memory");
        }
        __syncthreads();
        const float* Xb = Xs[cur];
        const float* Wb = Ws[cur];
        #pragma unroll
        for (int kk = 0; kk < 32; kk += 4) {
            v2f av, bv0, bv1;
            av.x  = Xb[arow + kk + khalf];
            av.y  = Xb[arow + kk + khalf + 1];
            bv0.x = Wb[brow0 + kk + khalf];
            bv0.y = Wb[brow0 + kk + khalf + 1];
            bv1.x = Wb[brow1 + kk + khalf];
            bv1.y = Wb[brow1 + kk + khalf + 1];
            acc0 = __builtin_amdgcn_wmma_f32_16x16x4_f32(
                false, av, false, bv0, (short)0, acc0, false, false);
            acc1 = __builtin_amdgcn_wmma_f32_16x16x4_f32(
                false, av, false, bv1, (short)0, acc1, false, false);
        }
        __syncthreads();
        cur ^= 1;
    }

    #pragma unroll
    for (int i = 0; i < 8; ++i) {
        int m = i + 8 * (lane >> 4);
        int s = rowToSample[m0 + m];
        if (s >= 0) {
            if (nIdx0 < NA) out[(size_t)s * NA + nIdx0] = acc0[i];
            if (nIdx1 < NA) out[(size_t)s * NA + nIdx1] = acc1[i];
        }
    }
}

extern "C" void kernel_launch(void* const* d_in, const int* in_sizes, int n_in,
                              void* d_out, int out_size, void* d_ws, size_t ws_size,
                              hipStream_t stream) {
    const float* mask = (const float*)d_in[0];   // [B,1,H,W]
    const float* feat = (const float*)d_in[1];   // [B,C,H,W]
    const float* W    = (const float*)d_in[2];   // [E,A,C]
    const float* bias = (const float*)d_in[3];   // [E,A]
    const int*   inst = (const int*)d_in[4];     // [B]
    float* out = (float*)d_out;                  // [B,A] fp32

    float* Xg          = (float*)d_ws;                              // [512,2048]
    int*   rowToSample = (int*)((char*)d_ws + (size_t)NROWS * NC * 4);
    int*   tileExpert  = rowToSample + NROWS;

    bucket_kernel<<<1, 1, 0, stream>>>(inst, rowToSample, tileExpert);
    pool_kernel<<<NROWS, 256, 0, stream>>>(mask, feat, rowToSample, Xg);
    gemm_kernel<<<dim3((NA + 127) / 128, NTILES), 128, 0, stream>>>(
        Xg, W, bias, rowToSample, tileExpert, out);
}